// Atten_81570018886343
// MI455X (gfx1250) — compile-verified
//
#include <hip/hip_runtime.h>

// Problem constants (B,T,K,C,H,D) = (8,16,49,2048,512,512)
#define BB 8
#define TT 16
#define KK 49
#define CC 2048
#define HH 512
#define DD 512
#define BT 128   // B*T

typedef float v2f __attribute__((ext_vector_type(2)));
typedef float v8f __attribute__((ext_vector_type(8)));

#define L2E 1.4426950408889634f

__device__ __forceinline__ float fast_exp(float x) {
  return __builtin_amdgcn_exp2f(x * L2E);
}

__device__ __forceinline__ float fast_tanh(float x) {
#if __has_builtin(__builtin_amdgcn_tanhf)
  return __builtin_amdgcn_tanhf(x);   // v_tanh_f32 (gfx1250 TRANS op)
#else
  x = fminf(fmaxf(x, -10.f), 10.f);
  float e = __builtin_amdgcn_exp2f(x * (2.0f * L2E));
  return (e - 1.0f) * __builtin_amdgcn_rcpf(e + 1.0f);
#endif
}

__device__ __forceinline__ float fast_sigmoid(float x) {
  float e = __builtin_amdgcn_exp2f(-x * L2E);
  return __builtin_amdgcn_rcpf(1.0f + e);
}

__device__ __forceinline__ v8f wmma_f32(v2f a, v2f b, v8f c) {
  // D = A(16x4) * B(4x16) + C ; f32 everywhere
  return __builtin_amdgcn_wmma_f32_16x16x4_f32(
      /*neg_a=*/false, a, /*neg_b=*/false, b,
      /*c_mod=*/(short)0, c, /*reuse_a=*/false, /*reuse_b=*/false);
}

// ---------------------------------------------------------------------------
// 1) featmean[b,c] = mean_k V[b,k,c]
// ---------------------------------------------------------------------------
__global__ __launch_bounds__(256) void featmean_kernel(
    const float* __restrict__ V, float* __restrict__ fm) {
  int b = blockIdx.x;
  for (int c = threadIdx.x; c < CC; c += 256) {
    float s = 0.f;
    const float* vb = V + (size_t)b * KK * CC + c;
    for (int k = 0; k < KK; ++k) s += vb[(size_t)k * CC];
    fm[b * CC + c] = s * (1.0f / (float)KK);
  }
}

// ---------------------------------------------------------------------------
// 2) Generic (dual-input) WMMA-f32 GEMM:  out = act( X1@W1^T [+ X2@W2^T] )
//    X*: M x H row-major, W*: N x H row-major, out: M x N row-major.
//    One wave per 16x16 output tile, K-steps of 4 (V_WMMA_F32_16X16X4_F32).
// ---------------------------------------------------------------------------
__global__ __launch_bounds__(256) void gemm_dual_wmma(
    const float* __restrict__ X1, const float* __restrict__ W1,
    const float* __restrict__ X2, const float* __restrict__ W2,
    float* __restrict__ out, int M, int N, int H, int act) {
  const int tmCnt = (M + 15) >> 4;
  const int tnCnt = (N + 15) >> 4;
  const int wid = blockIdx.x * 8 + (threadIdx.x >> 5);
  if (wid >= tmCnt * tnCnt) return;  // wave-uniform exit
  const int tM = wid / tnCnt, tN = wid % tnCnt;
  const int lane = threadIdx.x & 31;
  const int half = lane >> 4, l15 = lane & 15;

  const int m = tM * 16 + l15;
  const int n = tN * 16 + l15;
  const float amask = (m < M) ? 1.f : 0.f;
  const float bmask = (n < N) ? 1.f : 0.f;
  const int mc = (m < M) ? m : (M - 1);
  const int nc = (n < N) ? n : (N - 1);

  v8f acc = {0.f, 0.f, 0.f, 0.f, 0.f, 0.f, 0.f, 0.f};

  {
    const float* xr = X1 + (size_t)mc * H + half * 2;
    const float* wr = W1 + (size_t)nc * H + half * 2;
    for (int h = 0; h < H; h += 4) {
      v2f a = *(const v2f*)(xr + h);
      v2f b = *(const v2f*)(wr + h);
      a.x *= amask; a.y *= amask;
      b.x *= bmask; b.y *= bmask;
      acc = wmma_f32(a, b, acc);
    }
  }
  if (X2 != nullptr) {
    const float* xr = X2 + (size_t)mc * H + half * 2;
    const float* wr = W2 + (size_t)nc * H + half * 2;
    for (int h = 0; h < H; h += 4) {
      v2f a = *(const v2f*)(xr + h);
      v2f b = *(const v2f*)(wr + h);
      a.x *= amask; a.y *= amask;
      b.x *= bmask; b.y *= bmask;
      acc = wmma_f32(a, b, acc);
    }
  }

#pragma unroll
  for (int j = 0; j < 8; ++j) {
    int mm = tM * 16 + j + half * 8;
    int nn = tN * 16 + l15;
    if (mm < M && nn < N) {
      float v = acc[j];
      if (act == 1) v = fast_sigmoid(v);
      out[(size_t)mm * N + nn] = v;
    }
  }
}

// ---------------------------------------------------------------------------
// 3) cxt[bt,c] = sum_d tanh(featmean[b,c]*Wfeat[d] + g2[bt,d]) * Wcxt[d]
//    Dominant kernel: 134M tanh. One (bt, 256-wide c-chunk) per block.
// ---------------------------------------------------------------------------
__global__ __launch_bounds__(256) void cxt_kernel(
    const float* __restrict__ fm, const float* __restrict__ g2,
    const float* __restrict__ Wfeat, const float* __restrict__ Wcxt,
    float* __restrict__ cxt) {
  __shared__ float g2s[DD], wfs[DD], wcs[DD];
  const int blk = blockIdx.x;          // 0 .. BT*8-1
  const int bt = blk >> 3, chunk = blk & 7;
  const int b = bt / TT;
  const int tid = threadIdx.x;
  for (int i = tid; i < DD; i += 256) {
    g2s[i] = g2[bt * DD + i];
    wfs[i] = Wfeat[i];
    wcs[i] = Wcxt[i];
  }
  __syncthreads();
  const int c = chunk * 256 + tid;
  const float f = fm[b * CC + c];
  float acc = 0.f;
  for (int d = 0; d < DD; ++d) {
    float x = fmaf(f, wfs[d], g2s[d]);
    acc = fmaf(fast_tanh(x), wcs[d], acc);
  }
  cxt[(size_t)bt * CC + c] = acc;
}

// ---------------------------------------------------------------------------
// 4) In-place softmax over C for each (b,t) row:  cxt -> alpha0
// ---------------------------------------------------------------------------
__global__ __launch_bounds__(256) void softmax_c_kernel(float* __restrict__ cxt) {
  __shared__ float red[256];
  const int bt = blockIdx.x, tid = threadIdx.x;
  float* row = cxt + (size_t)bt * CC;
  float v[8];
  float mx = -1e30f;
#pragma unroll
  for (int i = 0; i < 8; ++i) { v[i] = row[tid + i * 256]; mx = fmaxf(mx, v[i]); }
  red[tid] = mx; __syncthreads();
  for (int o = 128; o > 0; o >>= 1) {
    if (tid < o) red[tid] = fmaxf(red[tid], red[tid + o]);
    __syncthreads();
  }
  mx = red[0]; __syncthreads();
  float s = 0.f;
#pragma unroll
  for (int i = 0; i < 8; ++i) { v[i] = fast_exp(v[i] - mx); s += v[i]; }
  red[tid] = s; __syncthreads();
  for (int o = 128; o > 0; o >>= 1) {
    if (tid < o) red[tid] += red[tid + o];
    __syncthreads();
  }
  const float inv = __builtin_amdgcn_rcpf(red[0]);
#pragma unroll
  for (int i = 0; i < 8; ++i) row[tid + i * 256] = v[i] * inv;
}

// ---------------------------------------------------------------------------
// 5) Per-(b,t) attention core. Block = 512 threads = 16 waves.
//    - WMMA: M[k,a] = sum_c (alpha0[c]*V[b,k,c]) * Wv[a,c]  (49x2048 @ 2048x49)
//    - z_t[k] = sum_a tanh(tanh(M[k,a]+g[bt,a]))*Wh[a]; alpha_t = softmax_K
//    - c_sp[c] = sum_k alpha_t[k] V[b,k,c]; c_ch[k] = mean_c alpha0[c]V[b,k,c]
//    - beta from extended softmax over [z_t, z_ext]
// ---------------------------------------------------------------------------
__global__ __launch_bounds__(512) void atten_core_kernel(
    const float* __restrict__ V, const float* __restrict__ Wv,
    const float* __restrict__ Wh, const float* __restrict__ alpha0,
    const float* __restrict__ g, const float* __restrict__ ssW,
    float* __restrict__ c_sp, float* __restrict__ c_ch,
    float* __restrict__ out_alpha, float* __restrict__ out_beta) {
  __shared__ float a0s[CC];
  __shared__ float Mt[64 * 64];
  __shared__ float gs[64], whs[64], sss[64], zts[64], ats[64], chs[64];
  __shared__ float sc[2];

  const int bt = blockIdx.x;
  const int b = bt / TT;
  const int tid = threadIdx.x;

  for (int c = tid; c < CC; c += 512) a0s[c] = alpha0[(size_t)bt * CC + c];
  if (tid < 64) {
    gs[tid]  = (tid < KK) ? g[bt * KK + tid]   : 0.f;
    whs[tid] = (tid < KK) ? Wh[tid]            : 0.f;
    sss[tid] = (tid < KK) ? ssW[bt * KK + tid] : 0.f;
  }
  __syncthreads();

  const int wave = tid >> 5, lane = tid & 31;
  const int half = lane >> 4, l15 = lane & 15;
  const int tM = wave >> 2, tN = wave & 3;   // 4x4 tiles of 16 covering 64x64

  // A operand row (k of P = alpha0 * V), B operand row (a of Wv)
  const int krow = tM * 16 + l15;
  const int arow = tN * 16 + l15;
  const float kmask = (krow < KK) ? 1.f : 0.f;
  const float amask = (arow < KK) ? 1.f : 0.f;
  const float* Vrow = V + (size_t)b * KK * CC +
                      (size_t)((krow < KK) ? krow : (KK - 1)) * CC + half * 2;
  const float* Wrow = Wv + (size_t)((arow < KK) ? arow : (KK - 1)) * CC + half * 2;

  v8f acc = {0.f, 0.f, 0.f, 0.f, 0.f, 0.f, 0.f, 0.f};
  for (int c = 0; c < CC; c += 4) {
    v2f a = *(const v2f*)(Vrow + c);
    v2f bb = *(const v2f*)(Wrow + c);
    const int c2 = c + half * 2;
    a.x *= a0s[c2] * kmask;
    a.y *= a0s[c2 + 1] * kmask;
    bb.x *= amask; bb.y *= amask;
    acc = wmma_f32(a, bb, acc);
  }
#pragma unroll
  for (int j = 0; j < 8; ++j)
    Mt[(tM * 16 + j + half * 8) * 64 + tN * 16 + l15] = acc[j];

  // channel sums: wave w handles k = w, w+16, w+32(, 48)
  for (int k = wave; k < KK; k += 16) {
    const float* vr = V + (size_t)b * KK * CC + (size_t)k * CC;
    float s = 0.f;
    for (int c = lane; c < CC; c += 32) s += a0s[c] * vr[c];
    for (int o = 16; o > 0; o >>= 1) s += __shfl_xor(s, o, 32);
    if (lane == 0) chs[k] = s * (1.0f / (float)CC);
  }
  __syncthreads();

  // z_t[k] = sum_a tanh(tanh(M[k,a] + g[a])) * Wh[a]   (g broadcasts over k!)
  if (tid < KK) {
    float z = 0.f;
    for (int a = 0; a < KK; ++a)
      z += fast_tanh(fast_tanh(Mt[tid * 64 + a] + gs[a])) * whs[a];
    zts[tid] = z;
  }
  __syncthreads();

  if (tid == 0) {
    float ze = 0.f;
    for (int k = 0; k < KK; ++k) ze += fast_tanh(sss[k]) * whs[k];
    float mx = -1e30f;
    for (int k = 0; k < KK; ++k) mx = fmaxf(mx, zts[k]);
    float sum = 0.f;
    for (int k = 0; k < KK; ++k) sum += fast_exp(zts[k] - mx);
    sc[0] = mx;
    sc[1] = __builtin_amdgcn_rcpf(sum);
    // extended softmax over [z_t(49), z_ext] -> beta = last element
    float mx2 = fmaxf(mx, ze);
    float eze = fast_exp(ze - mx2);
    float den = eze;
    for (int k = 0; k < KK; ++k) den += fast_exp(zts[k] - mx2);
    out_beta[bt] = eze / den;
  }
  __syncthreads();

  if (tid < KK) {
    float at = fast_exp(zts[tid] - sc[0]) * sc[1];
    ats[tid] = at;
    out_alpha[bt * KK + tid] = at;
    c_ch[bt * KK + tid] = chs[tid];
  }
  __syncthreads();

  // c_t_spatial[c] = sum_k alpha_t[k] * V[b,k,c]
  const float* vb = V + (size_t)b * KK * CC;
  for (int c = tid; c < CC; c += 512) {
    float s = 0.f;
    for (int k = 0; k < KK; ++k) s = fmaf(ats[k], vb[(size_t)k * CC + c], s);
    c_sp[(size_t)bt * CC + c] = s;
  }
}

// ---------------------------------------------------------------------------
// 6) c_hat[bt,d] = s_gate*spatial_info + c_gate*(sum_k c_ch[k]*Wch[d,k])
// ---------------------------------------------------------------------------
__global__ __launch_bounds__(256) void final_kernel(
    const float* __restrict__ sg, const float* __restrict__ cg,
    const float* __restrict__ spat, const float* __restrict__ cch,
    const float* __restrict__ Wch, float* __restrict__ out) {
  const int idx = blockIdx.x * 256 + threadIdx.x;   // BT*DD = 65536
  if (idx >= BT * DD) return;
  const int bt = idx >> 9;      // /512
  const int d = idx & 511;
  const float* cc = cch + bt * KK;
  const float* w = Wch + (size_t)d * KK;
  float ch = 0.f;
  for (int k = 0; k < KK; ++k) ch = fmaf(cc[k], w[k], ch);
  out[idx] = sg[idx] * spat[idx] + cg[idx] * ch;
}

// ---------------------------------------------------------------------------
static inline void launch_gemm(const float* X1, const float* W1,
                               const float* X2, const float* W2, float* out,
                               int M, int N, int H, int act, hipStream_t s) {
  int waves = ((M + 15) >> 4) * ((N + 15) >> 4);
  int blocks = (waves + 7) / 8;
  gemm_dual_wmma<<<blocks, 256, 0, s>>>(X1, W1, X2, W2, out, M, N, H, act);
}

extern "C" void kernel_launch(void* const* d_in, const int* in_sizes, int n_in,
                              void* d_out, int out_size, void* d_ws, size_t ws_size,
                              hipStream_t stream) {
  const float* V     = (const float*)d_in[0];
  const float* h_t   = (const float*)d_in[1];
  const float* s_t   = (const float*)d_in[2];
  const float* Wv    = (const float*)d_in[3];
  const float* Wg    = (const float*)d_in[4];
  const float* Ws    = (const float*)d_in[5];
  const float* Wh    = (const float*)d_in[6];
  const float* Wfeat = (const float*)d_in[7];
  const float* Wcxt  = (const float*)d_in[8];
  const float* Wg2   = (const float*)d_in[9];
  const float* Wsp   = (const float*)d_in[10];
  const float* Wch   = (const float*)d_in[11];
  const float* Wvs   = (const float*)d_in[12];
  const float* Wvc   = (const float*)d_in[13];
  const float* Whs   = (const float*)d_in[14];
  const float* Whc   = (const float*)d_in[15];

  float* ws = (float*)d_ws;
  float* fm   = ws;             // 16384  : featmean (B,C)
  float* g2   = ws + 16384;     // 65536  : g2 (BT,D)     (reused as spat)
  float* g    = ws + 81920;     // 6272   : g (BT,K)
  float* ssW  = ws + 88192;     // 6272   : content_s (BT,K)
  float* sg   = ws + 94464;     // 65536  : s_gate (BT,D)
  float* cg   = ws + 160000;    // 65536  : c_gate (BT,D)
  float* cxt  = ws + 225536;    // 262144 : cxt -> alpha0 in place (BT,C)
  float* csp  = ws + 487680;    // 262144 : c_t_spatial (BT,C)
  float* cch  = ws + 749824;    // 6272   : c_t_channel (BT,K)
  float* spat = g2;             // reuse g2 slot after cxt_kernel consumed it

  float* out_chat  = (float*)d_out;        // (B,T,D)  65536
  float* out_alpha = out_chat + BT * DD;   // (B,T,K)  6272
  float* out_beta  = out_alpha + BT * KK;  // (B,T,1)  128

  featmean_kernel<<<BB, 256, 0, stream>>>(V, fm);

  // projections (all WMMA f32)
  launch_gemm(h_t, Wg2, nullptr, nullptr, g2, BT, DD, HH, 0, stream);        // g2
  launch_gemm(h_t, Wg,  nullptr, nullptr, g,  BT, KK, HH, 0, stream);        // g
  launch_gemm(s_t, Ws,  h_t, Wg,          ssW, BT, KK, HH, 0, stream);       // content_s
  launch_gemm(s_t, Wvs, h_t, Whs,         sg,  BT, DD, HH, 1, stream);       // s_gate
  launch_gemm(s_t, Wvc, h_t, Whc,         cg,  BT, DD, HH, 1, stream);       // c_gate

  cxt_kernel<<<BT * 8, 256, 0, stream>>>(fm, g2, Wfeat, Wcxt, cxt);
  softmax_c_kernel<<<BT, 256, 0, stream>>>(cxt);  // cxt -> alpha0 in place

  atten_core_kernel<<<BT, 512, 0, stream>>>(V, Wv, Wh, cxt, g, ssW,
                                            csp, cch, out_alpha, out_beta);

  launch_gemm(csp, Wsp, nullptr, nullptr, spat, BT, DD, CC, 0, stream);      // spatial_info
  final_kernel<<<(BT * DD + 255) / 256, 256, 0, stream>>>(sg, cg, spat, cch,
                                                          Wch, out_chat);
}